// TransitionDown_70385924046904
// MI455X (gfx1250) — compile-verified
//
#include <hip/hip_runtime.h>

#define B_    8
#define N_    8192
#define S_    2048
#define KNN_  16
#define D_    128
#define COUT_ 256
#define CINP_ 132      // 131 channels padded to 132 (row 131 of w1f is zero)
#define X2STRIDE 260   // 256 + 4 pad -> stride mod 64 banks == 4 -> conflict-free frags

typedef float v2f __attribute__((ext_vector_type(2)));
typedef float v8f __attribute__((ext_vector_type(8)));

// ---------- wave32 reductions ----------
__device__ __forceinline__ void lexmin32(float& d, int& i) {
#pragma unroll
  for (int off = 16; off > 0; off >>= 1) {
    float od = __shfl_xor(d, off, 32);
    int   oi = __shfl_xor(i, off, 32);
    if (od < d || (od == d && oi < i)) { d = od; i = oi; }
  }
}

__device__ __forceinline__ void argmax32(float& v, int& i) {
#pragma unroll
  for (int off = 16; off > 0; off >>= 1) {
    float ov = __shfl_xor(v, off, 32);
    int   oi = __shfl_xor(i, off, 32);
    if (ov > v || (ov == v && oi < i)) { v = ov; i = oi; }
  }
}

// ---------- kernel 1: fold eval-mode BN into weights/bias ----------
__global__ void prep_weights(const float* __restrict__ w1, const float* __restrict__ b1,
                             const float* __restrict__ g1, const float* __restrict__ be1,
                             const float* __restrict__ m1, const float* __restrict__ v1,
                             const float* __restrict__ w2, const float* __restrict__ b2,
                             const float* __restrict__ g2, const float* __restrict__ be2,
                             const float* __restrict__ m2, const float* __restrict__ v2,
                             float* __restrict__ w1f, float* __restrict__ w2f,
                             float* __restrict__ bias1, float* __restrict__ bias2) {
  const int TOT1 = CINP_ * COUT_;
  const int TOT2 = COUT_ * COUT_;
  int total = TOT1 + TOT2 + 2 * COUT_;
  for (int t = blockIdx.x * blockDim.x + threadIdx.x; t < total; t += gridDim.x * blockDim.x) {
    if (t < TOT1) {
      int c = t / COUT_, o = t % COUT_;
      float sc = g1[o] * rsqrtf(v1[o] + 1e-5f);
      float w;
      if (c < 128)      w = w1[(c + 3) * COUT_ + o];   // feature channels (ref cols 3..130)
      else if (c < 131) w = w1[(c - 128) * COUT_ + o]; // xyz channels (ref cols 0..2)
      else              w = 0.f;                       // pad row
      w1f[t] = w * sc;
    } else if (t < TOT1 + TOT2) {
      int e = t - TOT1;
      int o = e % COUT_;
      float sc = g2[o] * rsqrtf(v2[o] + 1e-5f);
      w2f[e] = w2[e] * sc;
    } else if (t < TOT1 + TOT2 + COUT_) {
      int o = t - (TOT1 + TOT2);
      float sc = g1[o] * rsqrtf(v1[o] + 1e-5f);
      bias1[o] = (b1[o] - m1[o]) * sc + be1[o];
    } else {
      int o = t - (TOT1 + TOT2 + COUT_);
      float sc = g2[o] * rsqrtf(v2[o] + 1e-5f);
      bias2[o] = (b2[o] - m2[o]) * sc + be2[o];
    }
  }
}

// ---------- kernel 2: farthest point sampling (1 block / batch, cloud in LDS) ----------
__global__ void __launch_bounds__(1024) fps_kernel(const float* __restrict__ xyz,
                                                   float* __restrict__ newxyz) {
  extern __shared__ float sm[];
  float* sx = sm;                    // 3*N_
  float* rv = sm + 3 * N_;           // 32 wave maxima
  int*   ri = (int*)(rv + 32);       // 32 wave argmax
  float* bc = (float*)(ri + 32);     // broadcast centroid (3)

  int b = blockIdx.x, tid = threadIdx.x;
  const float* xb = xyz + (size_t)b * N_ * 3;
  for (int i = tid; i < 3 * N_; i += 1024) sx[i] = xb[i];
  __syncthreads();
  if (tid == 0) { bc[0] = sx[0]; bc[1] = sx[1]; bc[2] = sx[2]; }
  __syncthreads();

  float dist[8];
#pragma unroll
  for (int j = 0; j < 8; ++j) dist[j] = 1e10f;

  int lane = tid & 31, wave = tid >> 5;
  for (int s = 0; s < S_; ++s) {
    float cx = bc[0], cy = bc[1], cz = bc[2];
    if (tid == 0) {
      float* o = newxyz + ((size_t)b * S_ + s) * 3;
      o[0] = cx; o[1] = cy; o[2] = cz;
    }
    float bv = -1.f; int bi = 0;
#pragma unroll
    for (int j = 0; j < 8; ++j) {
      int n = tid + j * 1024;
      float dx = sx[3 * n] - cx, dy = sx[3 * n + 1] - cy, dz = sx[3 * n + 2] - cz;
      float d = dx * dx + dy * dy + dz * dz;
      d = fminf(dist[j], d);
      dist[j] = d;
      if (d > bv || (d == bv && n < bi)) { bv = d; bi = n; }
    }
    argmax32(bv, bi);
    if (lane == 0) { rv[wave] = bv; ri[wave] = bi; }
    __syncthreads();
    if (wave == 0) {
      float v = rv[lane]; int i2 = ri[lane];
      argmax32(v, i2);
      if (lane == 0) { bc[0] = sx[3 * i2]; bc[1] = sx[3 * i2 + 1]; bc[2] = sx[3 * i2 + 2]; }
    }
    __syncthreads();
  }
}

// ---------- kernel 3: kNN (16 lexicographic-min passes, cloud in LDS) ----------
#define KNN_CHUNK 256
__global__ void __launch_bounds__(256) knn_kernel(const float* __restrict__ xyz,
                                                  const float* __restrict__ newxyz,
                                                  int* __restrict__ knn) {
  extern __shared__ float sx[]; // 3*N_
  int b = blockIdx.x / (S_ / KNN_CHUNK);
  int chunk = blockIdx.x % (S_ / KNN_CHUNK);
  const float* xb = xyz + (size_t)b * N_ * 3;
  for (int i = threadIdx.x; i < 3 * N_; i += 256) sx[i] = xb[i];
  __syncthreads();

  int lane = threadIdx.x & 31, wave = threadIdx.x >> 5;
  for (int c = 0; c < KNN_CHUNK / 8; ++c) {
    int s = chunk * KNN_CHUNK + c * 8 + wave;
    const float* q = newxyz + ((size_t)b * S_ + s) * 3;
    float qx = q[0], qy = q[1], qz = q[2];
    float prevd = -1.f; int previ = -1;
    for (int k = 0; k < KNN_; ++k) {
      float bd = 3.4e38f; int bi = 0x7fffffff;
      for (int i = 0; i < N_ / 32; ++i) {
        int n = lane + i * 32;
        float dx = sx[3 * n] - qx, dy = sx[3 * n + 1] - qy, dz = sx[3 * n + 2] - qz;
        float d = dx * dx + dy * dy + dz * dz;
        bool ok = (d > prevd) || (d == prevd && n > previ);
        if (ok && (d < bd || (d == bd && n < bi))) { bd = d; bi = n; }
      }
      lexmin32(bd, bi);
      if (lane == 0) knn[((size_t)(b * S_ + s)) * KNN_ + k] = bi;
      prevd = bd; previ = bi;
    }
  }
}

// ---------- kernel 4: gather + fp32-WMMA MLP (2 layers) + neighbor max-pool ----------
#define WAVES_D 8
#define CPW_D   4
__global__ void __launch_bounds__(256) mlp_kernel(const float* __restrict__ xyz,
                                                  const float* __restrict__ points,
                                                  const float* __restrict__ newxyz,
                                                  const int* __restrict__ knn,
                                                  const float* __restrict__ w1f,
                                                  const float* __restrict__ bias1,
                                                  const float* __restrict__ w2f,
                                                  const float* __restrict__ bias2,
                                                  float* __restrict__ newpoints) {
  extern __shared__ float sm[];
  int lane = threadIdx.x & 31, wave = threadIdx.x >> 5;
  float* A1 = sm + wave * (16 * CINP_ + 16 * X2STRIDE); // per-wave [16 x 132] input tile
  float* X2 = A1 + 16 * CINP_;                          // per-wave [16 x 260] layer-1 out
  int row = lane & 15, half = lane >> 4;

  for (int c = 0; c < CPW_D; ++c) {
    int id = blockIdx.x * (WAVES_D * CPW_D) + wave * CPW_D + c;  // b*S + s
    int b = id / S_;
    // --- gather: cols 0..127 = point feats, 128..130 = rel xyz, 131 = pad ---
    int nidx = knn[(size_t)id * KNN_ + row];
    const float* prow = points + ((size_t)b * N_ + nidx) * D_ + half * 64;
    float* a1r = A1 + row * CINP_ + half * 64;
#pragma unroll
    for (int j = 0; j < 16; ++j) {
      const float4 v = *(const float4*)(prow + 4 * j);
      a1r[4 * j + 0] = v.x; a1r[4 * j + 1] = v.y;
      a1r[4 * j + 2] = v.z; a1r[4 * j + 3] = v.w;
    }
    if (half == 0) {
      const float* xp = xyz + ((size_t)b * N_ + nidx) * 3;
      const float* q  = newxyz + (size_t)id * 3;
      A1[row * CINP_ + 128] = xp[0] - q[0];
      A1[row * CINP_ + 129] = xp[1] - q[1];
      A1[row * CINP_ + 130] = xp[2] - q[2];
    } else {
      A1[row * CINP_ + 131] = 0.f;
    }

    // --- layer 1: [16x132] x [132x256] via V_WMMA_F32_16X16X4_F32 ---
#pragma unroll
    for (int g = 0; g < 2; ++g) {
      v8f acc[8] = {};
      for (int k = 0; k < 33; ++k) {
        v2f a = *(const v2f*)(A1 + row * CINP_ + 4 * k + 2 * half);
        const float* wp = w1f + (4 * k + 2 * half) * COUT_ + g * 128 + row;
#pragma unroll
        for (int t = 0; t < 8; ++t) {
          v2f bf; bf.x = wp[t * 16]; bf.y = wp[COUT_ + t * 16];
          acc[t] = __builtin_amdgcn_wmma_f32_16x16x4_f32(
              false, a, false, bf, (short)0, acc[t], false, false);
        }
      }
#pragma unroll
      for (int t = 0; t < 8; ++t) {
        int n = g * 128 + t * 16 + row;
        float bias = bias1[n];
#pragma unroll
        for (int v = 0; v < 8; ++v) {
          float y = fmaxf(acc[t][v] + bias, 0.f);
          X2[(v + 8 * half) * X2STRIDE + n] = y;   // D-layout -> row-major LDS
        }
      }
    }

    // --- layer 2: [16x256] x [256x256], then max over the 16 neighbors ---
#pragma unroll
    for (int g = 0; g < 2; ++g) {
      v8f acc[8] = {};
      for (int k = 0; k < 64; ++k) {
        v2f a = *(const v2f*)(X2 + row * X2STRIDE + 4 * k + 2 * half);
        const float* wp = w2f + (4 * k + 2 * half) * COUT_ + g * 128 + row;
#pragma unroll
        for (int t = 0; t < 8; ++t) {
          v2f bf; bf.x = wp[t * 16]; bf.y = wp[COUT_ + t * 16];
          acc[t] = __builtin_amdgcn_wmma_f32_16x16x4_f32(
              false, a, false, bf, (short)0, acc[t], false, false);
        }
      }
#pragma unroll
      for (int t = 0; t < 8; ++t) {
        int n = g * 128 + t * 16 + row;
        float m = acc[t][0];
#pragma unroll
        for (int v = 1; v < 8; ++v) m = fmaxf(m, acc[t][v]);   // max over 8 rows this half
        float om = __shfl_xor(m, 16, 32);                       // combine lane halves
        m = fmaxf(m, om);
        float y = fmaxf(m + bias2[n], 0.f);                     // bias const in M; relu monotone
        if (half == 0) newpoints[(size_t)id * COUT_ + n] = y;
      }
    }
  }
}

// ---------- host ----------
extern "C" void kernel_launch(void* const* d_in, const int* in_sizes, int n_in,
                              void* d_out, int out_size, void* d_ws, size_t ws_size,
                              hipStream_t stream) {
  (void)in_sizes; (void)n_in; (void)out_size; (void)ws_size;
  const float* xyz = (const float*)d_in[0];
  const float* pts = (const float*)d_in[1];
  const float* w1  = (const float*)d_in[2];
  const float* b1  = (const float*)d_in[3];
  const float* g1  = (const float*)d_in[4];
  const float* be1 = (const float*)d_in[5];
  const float* m1  = (const float*)d_in[6];
  const float* v1  = (const float*)d_in[7];
  const float* w2  = (const float*)d_in[8];
  const float* b2  = (const float*)d_in[9];
  const float* g2  = (const float*)d_in[10];
  const float* be2 = (const float*)d_in[11];
  const float* m2  = (const float*)d_in[12];
  const float* v2  = (const float*)d_in[13];

  float* out_xyz = (float*)d_out;                       // [8,2048,3]
  float* out_pts = out_xyz + (size_t)B_ * S_ * 3;       // [8,2048,256]

  char* ws = (char*)d_ws;
  int*   knn   = (int*)ws;                                        // 1,048,576 B
  float* w1f   = (float*)(ws + 1048576);                          //   135,168 B
  float* w2f   = (float*)(ws + 1048576 + 135168);                 //   262,144 B
  float* bias1 = (float*)(ws + 1048576 + 135168 + 262144);        //     1,024 B
  float* bias2 = bias1 + COUT_;                                   //     1,024 B

  prep_weights<<<256, 256, 0, stream>>>(w1, b1, g1, be1, m1, v1,
                                        w2, b2, g2, be2, m2, v2,
                                        w1f, w2f, bias1, bias2);

  size_t fps_lds = (size_t)(3 * N_) * 4 + 32 * 4 + 32 * 4 + 16;
  fps_kernel<<<B_, 1024, fps_lds, stream>>>(xyz, out_xyz);

  size_t knn_lds = (size_t)(3 * N_) * 4;
  knn_kernel<<<B_ * (S_ / KNN_CHUNK), 256, knn_lds, stream>>>(xyz, out_xyz, knn);

  size_t mlp_lds = (size_t)WAVES_D * (16 * CINP_ + 16 * X2STRIDE) * 4;  // 200,704 B
  int mlp_blocks = (B_ * S_) / (WAVES_D * CPW_D);                        // 512
  mlp_kernel<<<mlp_blocks, 256, mlp_lds, stream>>>(xyz, pts, out_xyz, knn,
                                                   w1f, bias1, w2f, bias2, out_pts);
}